// GraphModelAttention_40948218200432
// MI455X (gfx1250) — compile-verified
//
#include <hip/hip_runtime.h>
#include <hip/hip_bf16.h>

#define N_NODES  50000
#define N_EDGES  800000
#define IN_CH    64
#define HID_CH   128
#define OUT_CH   64
#define N_GRAPHS 512

typedef float v2f __attribute__((ext_vector_type(2)));
typedef float v8f __attribute__((ext_vector_type(8)));

// D(16x16,f32) = A(16x4,f32) * B(4x16,f32) + C  -> v_wmma_f32_16x16x4_f32
__device__ __forceinline__ v8f wmma_f32_16x16x4(v2f a, v2f b, v8f c) {
  return __builtin_amdgcn_wmma_f32_16x16x4_f32(false, a, false, b, (short)0, c,
                                               false, false);
}

// ---------------------------------------------------------------------------
// Kernel 1: edge scatter — agg[dst] += x[src], deg[dst] += 1   (memory bound)
// 16 threads per edge, float4 gather + 4 f32 atomics each.
// ---------------------------------------------------------------------------
__global__ void __launch_bounds__(256)
edge_scatter_kernel(const float* __restrict__ x,
                    const long long* __restrict__ edge_index,
                    float* __restrict__ agg,
                    float* __restrict__ deg) {
  unsigned gid = blockIdx.x * blockDim.x + threadIdx.x;
  unsigned e   = gid >> 4;
  if (e >= N_EDGES) return;
  unsigned c4  = (gid & 15u) << 2;
  long long s  = edge_index[e];
  long long d  = edge_index[(size_t)N_EDGES + e];
  const float4 v = *(const float4*)(x + (size_t)s * IN_CH + c4);
  float* dp = agg + (size_t)d * IN_CH + c4;
  atomicAdd(dp + 0, v.x);
  atomicAdd(dp + 1, v.y);
  atomicAdd(dp + 2, v.z);
  atomicAdd(dp + 3, v.w);
  if (c4 == 0) atomicAdd(deg + d, 1.0f);
}

// ---------------------------------------------------------------------------
// Kernel 2: fused SAGEConv + ReLU + attention + global mean-pool (numerator).
// One wave per 16-node tile. h = relu((agg/deg) @ Wl^T + b_l + x @ Wr^T),
// s = sigmoid(h @ Watt^T + b_att), pooled[batch] += h*s, counts[batch] += 1.
// Wl/Wr staged in LDS; 256 v_wmma_f32_16x16x4_f32 per wave.
// ---------------------------------------------------------------------------
__global__ void __launch_bounds__(256)
sage_fused_kernel(const float* __restrict__ x,
                  const long long* __restrict__ batch,
                  const float* __restrict__ agg,
                  const float* __restrict__ deg,
                  const float* __restrict__ W_l,
                  const float* __restrict__ b_l,
                  const float* __restrict__ W_r,
                  const float* __restrict__ W_att,
                  const float* __restrict__ b_att,
                  float* __restrict__ pooled,
                  float* __restrict__ counts) {
  __shared__ float sWl[HID_CH * IN_CH];  // 32 KB
  __shared__ float sWr[HID_CH * IN_CH];  // 32 KB
  for (int i = threadIdx.x; i < (HID_CH * IN_CH) / 4; i += blockDim.x) {
    ((float4*)sWl)[i] = ((const float4*)W_l)[i];
    ((float4*)sWr)[i] = ((const float4*)W_r)[i];
  }
  __syncthreads();

  const int lane = threadIdx.x & 31;
  const int tile = blockIdx.x * (blockDim.x >> 5) + (threadIdx.x >> 5);
  const int nTiles = N_NODES / 16;  // 3125 (exact)
  if (tile >= nTiles) return;       // wave-uniform: EXEC stays all-1s for WMMA

  const int nodeBase = tile * 16;
  const int col  = lane & 15;  // N / M position
  const int half = lane >> 4;  // K-pair selector for A/B, row-half for C/D
  const int koff = half * 2;

  // --- cache A operands in VGPRs: one x-row and one (agg/deg)-row per lane ---
  const int rowA = nodeBase + col;  // A layout: lane l holds row l%16
  const float invd = 1.0f / fmaxf(deg[rowA], 1.0f);
  const float* xrow = x   + (size_t)rowA * IN_CH;
  const float* grow = agg + (size_t)rowA * IN_CH;
  v2f xa[16], ga[16];
#pragma unroll
  for (int k = 0; k < 16; ++k) {
    xa[k] = *(const v2f*)(xrow + k * 4 + koff);
    v2f g = *(const v2f*)(grow + k * 4 + koff);
    ga[k] = g * invd;  // mean aggregation folded into A
  }

  // --- 8 column tiles of 16 HID channels each ---
  v8f acc[8];
#pragma unroll
  for (int t = 0; t < 8; ++t) {
    const float bias = b_l[t * 16 + col];
    v8f c = {bias, bias, bias, bias, bias, bias, bias, bias};
    const float* wr = sWr + (t * 16 + col) * IN_CH + koff;  // B[k][n] = W[n][k]
    const float* wl = sWl + (t * 16 + col) * IN_CH + koff;
#pragma unroll
    for (int k = 0; k < 16; ++k) {  // x @ Wr^T
      v2f b = *(const v2f*)(wr + k * 4);
      c = wmma_f32_16x16x4(xa[k], b, c);
    }
#pragma unroll
    for (int k = 0; k < 16; ++k) {  // (agg/deg) @ Wl^T
      v2f b = *(const v2f*)(wl + k * 4);
      c = wmma_f32_16x16x4(ga[k], b, c);
    }
    float* cf = (float*)&c;
#pragma unroll
    for (int i = 0; i < 8; ++i) cf[i] = fmaxf(cf[i], 0.0f);  // ReLU
    acc[t] = c;
  }

  // --- attention: score[m] = sigmoid(h[m,:] . W_att + b_att) ---
  float watt[8];
#pragma unroll
  for (int t = 0; t < 8; ++t) watt[t] = W_att[t * 16 + col];
  const float batt = b_att[0];

#pragma unroll
  for (int r = 0; r < 8; ++r) {  // row m = nodeBase + r + half*8
    float s = 0.0f;
#pragma unroll
    for (int t = 0; t < 8; ++t) s += ((const float*)&acc[t])[r] * watt[t];
    // reduce over the 16 lanes of this half (xor 1,2,4,8 stays in-half)
#pragma unroll
    for (int off = 1; off < 16; off <<= 1) s += __shfl_xor(s, off, 32);
    const float score = 1.0f / (1.0f + __expf(-(s + batt)));
#pragma unroll
    for (int t = 0; t < 8; ++t) ((float*)&acc[t])[r] *= score;
  }

  // --- fused global mean-pool numerator + counts ---
#pragma unroll
  for (int r = 0; r < 8; ++r) {
    const int m = nodeBase + r + half * 8;  // C/D layout
    const long long g = batch[m];
    float* prow = pooled + (size_t)g * HID_CH;
#pragma unroll
    for (int t = 0; t < 8; ++t)
      atomicAdd(prow + t * 16 + col, ((const float*)&acc[t])[r]);
    if (col == 0) atomicAdd(counts + g, 1.0f);  // once per node (lanes 0 & 16)
  }
}

// ---------------------------------------------------------------------------
// Kernel 3: out = (pooled/counts) @ W_out^T + b_out   [512x128]@[128x64]
// One wave per 16x16 output tile; 32 WMMAs over K=128.
// ---------------------------------------------------------------------------
__global__ void __launch_bounds__(128)
out_gemm_kernel(const float* __restrict__ pooled,
                const float* __restrict__ counts,
                const float* __restrict__ W_out,
                const float* __restrict__ b_out,
                float* __restrict__ out) {
  const int lane = threadIdx.x & 31;
  const int tile = blockIdx.x * (blockDim.x >> 5) + (threadIdx.x >> 5);  // 0..127
  const int rt = tile >> 2;   // 32 row tiles
  const int ct = tile & 3;    // 4 col tiles
  const int col  = lane & 15;
  const int half = lane >> 4;
  const int koff = half * 2;

  const int rowA = rt * 16 + col;
  const float invc = 1.0f / fmaxf(counts[rowA], 1.0f);
  const float* arow = pooled + (size_t)rowA * HID_CH + koff;
  const float* brow = W_out + (size_t)(ct * 16 + col) * HID_CH + koff;

  const float bias = b_out[ct * 16 + col];
  v8f c = {bias, bias, bias, bias, bias, bias, bias, bias};
#pragma unroll
  for (int k = 0; k < 32; ++k) {
    v2f a = *(const v2f*)(arow + k * 4);
    a = a * invc;  // mean pool denominator folded into A
    v2f b = *(const v2f*)(brow + k * 4);
    c = wmma_f32_16x16x4(a, b, c);
  }
  const float* cf = (const float*)&c;
#pragma unroll
  for (int r = 0; r < 8; ++r)
    out[(size_t)(rt * 16 + r + half * 8) * OUT_CH + ct * 16 + col] = cf[r];
}

// ---------------------------------------------------------------------------
extern "C" void kernel_launch(void* const* d_in, const int* in_sizes, int n_in,
                              void* d_out, int out_size, void* d_ws,
                              size_t ws_size, hipStream_t stream) {
  const float*     x     = (const float*)d_in[0];
  const long long* ei    = (const long long*)d_in[1];  // int64 in reference
  const long long* batch = (const long long*)d_in[2];  // int64, sorted
  const float* W_l   = (const float*)d_in[3];
  const float* b_l   = (const float*)d_in[4];
  const float* W_r   = (const float*)d_in[5];
  const float* W_att = (const float*)d_in[6];
  const float* b_att = (const float*)d_in[7];
  const float* W_out = (const float*)d_in[8];
  const float* b_out = (const float*)d_in[9];
  float* out = (float*)d_out;

  // workspace layout (floats): agg | deg | pooled | counts  (~13.3 MB)
  float* ws     = (float*)d_ws;
  float* agg    = ws;                                    // 50000*64
  float* deg    = agg + (size_t)N_NODES * IN_CH;         // 50000
  float* pooled = deg + N_NODES;                         // 512*128
  float* counts = pooled + (size_t)N_GRAPHS * HID_CH;    // 512
  const size_t zeroFloats =
      (size_t)N_NODES * IN_CH + N_NODES + (size_t)N_GRAPHS * HID_CH + N_GRAPHS;
  hipMemsetAsync(d_ws, 0, zeroFloats * sizeof(float), stream);

  {  // edge scatter: 800000 edges * 16 threads
    unsigned threads = N_EDGES * 16u;
    edge_scatter_kernel<<<dim3((threads + 255u) / 256u), dim3(256), 0, stream>>>(
        x, ei, agg, deg);
  }
  {  // fused SAGE: 3125 node tiles, 8 waves per block
    int nTiles = N_NODES / 16;
    int wpb = 256 / 32;
    sage_fused_kernel<<<dim3((nTiles + wpb - 1) / wpb), dim3(256), 0, stream>>>(
        x, batch, agg, deg, W_l, b_l, W_r, W_att, b_att, pooled, counts);
  }
  {  // output GEMM: 128 tiles, 4 waves per block
    out_gemm_kernel<<<dim3(32), dim3(128), 0, stream>>>(pooled, counts, W_out,
                                                        b_out, out);
  }
}